// TDRumorGCN_5480378270219
// MI455X (gfx1250) — compile-verified
//
#include <hip/hip_runtime.h>

typedef __attribute__((ext_vector_type(2))) float v2f;
typedef __attribute__((ext_vector_type(8))) float v8f;

#define NGRAPH 128
#define FH     64   // HID == OUT == 64

// ---------------------------------------------------------------------------
// degree / normalization
// ---------------------------------------------------------------------------
__global__ void init_deg_kernel(float* deg, int n) {
    int i = blockIdx.x * blockDim.x + threadIdx.x;
    if (i < n) deg[i] = 1.0f;                       // self-loop
}

__global__ void edge_deg_kernel(const int* __restrict__ dst, float* deg, int e) {
    int i = blockIdx.x * blockDim.x + threadIdx.x;
    if (i < e) atomicAdd(&deg[dst[i]], 1.0f);
}

__global__ void rsqrt_kernel(float* deg, int n) {
    int i = blockIdx.x * blockDim.x + threadIdx.x;
    if (i < n) deg[i] = rsqrtf(deg[i]);             // deg >= 1 always
}

// ---------------------------------------------------------------------------
// WMMA f32 GEMM:  C[M,64] = (relu?) A[rows, K] @ B[K,64]
// Register-blocked: each wave computes 16 rows x 64 cols with 4 accumulators,
// so the A fragment is loaded ONCE per K-step and feeds 4 WMMAs (4x less A
// operand traffic than one-tile-per-wave).  Block = 256 thr = 8 waves
// covering 128 rows; M is always a multiple of 128 here (102400, 128).
// Fragment layout per CDNA5 ISA 7.12.2:
//   A (16x4):  lanes 0-15: M=lane, v0=K0,v1=K1; lanes 16-31: M=lane-16, v0=K2,v1=K3
//   B (4x16):  v0: K = k0+2*hi, N = lane&15 ; v1: next K row
//   C (16x16): vgpr v: M = v + 8*hi, N = lane&15
// ---------------------------------------------------------------------------
template <bool RELU_A>
__global__ __launch_bounds__(256)
void gemm_wmma_kernel(const float* __restrict__ A,
                      const int*   __restrict__ rowidx,
                      const float* __restrict__ B,
                      float*       __restrict__ C,
                      int M, int K)
{
    const int tid   = threadIdx.x;
    const int wave  = tid >> 5;
    const int lane  = tid & 31;
    const int r0    = (blockIdx.x * 8 + wave) * 16;
    if (r0 >= M) return;                            // wave-uniform

    const int mLane = lane & 15;
    const int hi    = lane >> 4;                    // 0 | 1
    const int row   = r0 + mLane;

    const long long abase = rowidx ? (long long)rowidx[row] * (long long)K
                                   : (long long)row * (long long)K;
    const float* __restrict__ Arow = A + abase + 2 * hi;
    const float* __restrict__ Bp   = B + (long long)(2 * hi) * FH + mLane;

    v8f acc0 = {}, acc1 = {}, acc2 = {}, acc3 = {};
    for (int k0 = 0; k0 < K; k0 += 4) {
        float a0 = Arow[k0];
        float a1 = Arow[k0 + 1];
        if (RELU_A) { a0 = fmaxf(a0, 0.0f); a1 = fmaxf(a1, 0.0f); }
        v2f a; a.x = a0; a.y = a1;

        const float* b0 = Bp + (long long)k0 * FH;       // K row k0+2*hi
        const float* b1 = b0 + FH;                       // next K row
        v2f bf0; bf0.x = b0[0];  bf0.y = b1[0];
        v2f bf1; bf1.x = b0[16]; bf1.y = b1[16];
        v2f bf2; bf2.x = b0[32]; bf2.y = b1[32];
        v2f bf3; bf3.x = b0[48]; bf3.y = b1[48];

        acc0 = __builtin_amdgcn_wmma_f32_16x16x4_f32(false, a, false, bf0, (short)0, acc0, false, false);
        acc1 = __builtin_amdgcn_wmma_f32_16x16x4_f32(false, a, false, bf1, (short)0, acc1, false, false);
        acc2 = __builtin_amdgcn_wmma_f32_16x16x4_f32(false, a, false, bf2, (short)0, acc2, false, false);
        acc3 = __builtin_amdgcn_wmma_f32_16x16x4_f32(false, a, false, bf3, (short)0, acc3, false, false);
    }

    float* Cp = C + (long long)(r0 + 8 * hi) * FH + mLane;
#pragma unroll
    for (int v = 0; v < 8; ++v) {
        Cp[(long long)v * FH +  0] = acc0[v];
        Cp[(long long)v * FH + 16] = acc1[v];
        Cp[(long long)v * FH + 32] = acc2[v];
        Cp[(long long)v * FH + 48] = acc3[v];
    }
}

// ---------------------------------------------------------------------------
// hs[n,f] = (h[n,f] + R[batch[n],f]) * dinv[n]      (R optional, in-place)
// ---------------------------------------------------------------------------
__global__ void scale_rows_kernel(float* h, const float* __restrict__ dinv,
                                  const float* __restrict__ R,
                                  const int* __restrict__ batch, int n)
{
    long long t = (long long)blockIdx.x * blockDim.x + threadIdx.x;
    if (t >= (long long)n * FH) return;
    int node = (int)(t >> 6), f = (int)(t & 63);
    float v = h[t];
    if (R) v += R[(long long)batch[node] * FH + f];
    h[t] = v * dinv[node];
}

// edge-parallel scatter: acc[dst] += hs[src]   (64 threads per edge)
__global__ void scatter_kernel(const int* __restrict__ src,
                               const int* __restrict__ dst,
                               const float* __restrict__ hs,
                               float* acc, int e)
{
    long long t = (long long)blockIdx.x * blockDim.x + threadIdx.x;
    if (t >= (long long)e * FH) return;
    int ed = (int)(t >> 6), f = (int)(t & 63);
    atomicAdd(&acc[(long long)dst[ed] * FH + f], hs[(long long)src[ed] * FH + f]);
}

// out[n,f] = (relu?)( dinv[n]*(acc[n,f] + hs[n,f]) + bias[f] )
template <bool RELU>
__global__ void finalize_kernel(const float* __restrict__ acc,
                                const float* __restrict__ hs,
                                const float* __restrict__ dinv,
                                const float* __restrict__ bias,
                                float* outp, int n)
{
    long long t = (long long)blockIdx.x * blockDim.x + threadIdx.x;
    if (t >= (long long)n * FH) return;
    int node = (int)(t >> 6), f = (int)(t & 63);
    float v = dinv[node] * (acc[t] + hs[t]) + bias[f];
    if (RELU) v = fmaxf(v, 0.0f);
    outp[t] = v;
}

// per-graph sums + counts
__global__ void graph_reduce_kernel(const float* __restrict__ h2,
                                    const int* __restrict__ batch,
                                    float* gsum, float* gcnt, int n)
{
    long long t = (long long)blockIdx.x * blockDim.x + threadIdx.x;
    if (t >= (long long)n * FH) return;
    int node = (int)(t >> 6), f = (int)(t & 63);
    int g = batch[node];
    atomicAdd(&gsum[(long long)g * FH + f], h2[t]);
    if (f == 0) atomicAdd(&gcnt[g], 1.0f);
}

// out[g,0:64] = gsum/max(cnt,1);  out[g,64:128] = cnt>0 ? x2[root[g]] : 0
__global__ void output_kernel(const float* __restrict__ gsum,
                              const float* __restrict__ gcnt,
                              const float* __restrict__ x2,
                              const int* __restrict__ root,
                              float* out)
{
    int t = blockIdx.x * blockDim.x + threadIdx.x;
    if (t >= NGRAPH * 2 * FH) return;
    int g = t >> 7, f = t & 127;
    float c = gcnt[g];
    float v;
    if (f < FH) v = gsum[g * FH + f] / fmaxf(c, 1.0f);
    else        v = (c > 0.0f) ? x2[(long long)root[g] * FH + (f - FH)] : 0.0f;
    out[t] = v;
}

// ---------------------------------------------------------------------------
extern "C" void kernel_launch(void* const* d_in, const int* in_sizes, int n_in,
                              void* d_out, int out_size, void* d_ws, size_t ws_size,
                              hipStream_t stream)
{
    const float* x     = (const float*)d_in[0];     // [N, 768]
    const int*   edge  = (const int*)  d_in[1];     // [2, E]
    const int*   batch = (const int*)  d_in[2];     // [N]
    const int*   root  = (const int*)  d_in[3];     // [G]
    const float* W1    = (const float*)d_in[4];     // [768, 64]
    const float* b1    = (const float*)d_in[5];     // [64]
    const float* W2    = (const float*)d_in[6];     // [832, 64]
    const float* b2    = (const float*)d_in[7];     // [64]
    float*       out   = (float*)d_out;             // [128, 128]

    const int N  = in_sizes[2];
    const int E  = in_sizes[1] / 2;
    const int G  = in_sizes[3];
    const int K1 = in_sizes[0] / N;                 // 768

    const int* srcI = edge;
    const int* dstI = edge + E;

    // workspace carve-up (floats)
    float* ws   = (float*)d_ws;
    float* dinv = ws;                               // N          (deg -> dinv)
    float* bufA = dinv + N;                         // N*64  (h1/hs, later h2pre/hs2/h2)
    float* acc  = bufA + (size_t)N * FH;            // N*64  (scatter accumulator)
    float* x2   = acc  + (size_t)N * FH;            // N*64  (conv1 output)
    float* R    = x2   + (size_t)N * FH;            // G*64  (root term of conv2)
    float* gsum = R    + (size_t)G * FH;            // G*64
    float* gcnt = gsum + (size_t)G * FH;            // G

    const int TB = 256;
    const long long nf = (long long)N * FH;
    const long long ef = (long long)E * FH;
    const int gNF = (int)((nf + TB - 1) / TB);
    const int gEF = (int)((ef + TB - 1) / TB);

    // --- degrees -> dinv ---
    init_deg_kernel<<<(N + TB - 1) / TB, TB, 0, stream>>>(dinv, N);
    edge_deg_kernel<<<(E + TB - 1) / TB, TB, 0, stream>>>(dstI, dinv, E);
    rsqrt_kernel<<<(N + TB - 1) / TB, TB, 0, stream>>>(dinv, N);

    // --- conv1: h1 = x @ W1 (WMMA f32), scale, scatter, finalize -> x2 ---
    gemm_wmma_kernel<false><<<N / 128, 256, 0, stream>>>(x, nullptr, W1, bufA, N, K1);
    scale_rows_kernel<<<gNF, TB, 0, stream>>>(bufA, dinv, nullptr, batch, N);
    hipMemsetAsync(acc, 0, (size_t)N * FH * sizeof(float), stream);
    scatter_kernel<<<gEF, TB, 0, stream>>>(srcI, dstI, bufA, acc, E);
    finalize_kernel<false><<<gNF, TB, 0, stream>>>(acc, bufA, dinv, b1, x2, N);

    // --- per-graph root term: R = relu(x[root]) @ W2[64:832] (tiny WMMA) ---
    gemm_wmma_kernel<true><<<(G + 127) / 128, 256, 0, stream>>>(x, root, W2 + (size_t)FH * FH, R, G, K1);

    // --- conv2: h2pre = relu(x2) @ W2[0:64]; + R[batch]; scale; scatter ---
    gemm_wmma_kernel<true><<<N / 128, 256, 0, stream>>>(x2, nullptr, W2, bufA, N, FH);
    scale_rows_kernel<<<gNF, TB, 0, stream>>>(bufA, dinv, R, batch, N);
    hipMemsetAsync(acc, 0, (size_t)N * FH * sizeof(float), stream);
    scatter_kernel<<<gEF, TB, 0, stream>>>(srcI, dstI, bufA, acc, E);
    finalize_kernel<true><<<gNF, TB, 0, stream>>>(acc, bufA, dinv, b2, bufA, N);  // h2 (relu'd) in place

    // --- per-graph mean + root concat ---
    hipMemsetAsync(gsum, 0, (size_t)(G * FH + G) * sizeof(float), stream);
    graph_reduce_kernel<<<gNF, TB, 0, stream>>>(bufA, batch, gsum, gcnt, N);
    output_kernel<<<(G * 2 * FH + TB - 1) / TB, TB, 0, stream>>>(gsum, gcnt, x2, root, out);
}